// Label_Distribution_Estimation_45689862095301
// MI455X (gfx1250) — compile-verified
//
#include <hip/hip_runtime.h>
#include <math.h>

// ---------------------------------------------------------------------------
// Label Distribution Estimation, optimized for MI455X (gfx1250, wave32).
// Key algebraic optimization: scores are only needed at the K=8 top-neighbor
// positions per row (mask multiply in the reference), cutting the dominant
// pair-MLP GEMM from 65536 pairs to 2048 pairs (32x).
// fp32 WMMA (v_wmma_f32_16x16x4_f32) everywhere; problem is BW-bound.
// ---------------------------------------------------------------------------

typedef float v2f __attribute__((ext_vector_type(2)));
typedef float v8f __attribute__((ext_vector_type(8)));

#define BDIM 256
#define FERD 2048
#define LMD  256
#define NC   7
#define KNN  8
#define EPSV 1e-8f
#define MOM  0.9f

// ---------------- row L2-normalize (for cosine similarity) ----------------
__global__ void row_normalize(const float* __restrict__ x,
                              float* __restrict__ out, int D) {
  const int r = blockIdx.x;
  const float* row = x + (size_t)r * D;
  float* orow = out + (size_t)r * D;
  __shared__ float red[256];
  float s = 0.f;
  for (int c = threadIdx.x; c < D; c += blockDim.x) {
    float v = row[c];
    s += v * v;
  }
  red[threadIdx.x] = s;
  __syncthreads();
  for (int off = 128; off > 0; off >>= 1) {
    if ((int)threadIdx.x < off) red[threadIdx.x] += red[threadIdx.x + off];
    __syncthreads();
  }
  const float inv = 1.0f / sqrtf(red[0]);
  for (int c = threadIdx.x; c < D; c += blockDim.x) orow[c] = row[c] * inv;
}

// ---------------- generic C = A * B^T (+bias) via fp32 WMMA ----------------
// A: M x Kd (row stride lda), Bsrc: N x Kd (row stride ldb) -> C: M x N.
// One wave per 16x16 output tile. CDNA5 fragment layouts:
//  A 16x4 : lane m = lane&15, K = 2*(lane>>4) + {0,1} in frag components
//  B 4x16 : lane n = lane&15, same K split
//  C 16x16: vgpr v -> M = v (lanes 0-15) / v+8 (lanes 16-31), N = lane&15
__global__ void gemm_nt_wmma(const float* __restrict__ A, int lda,
                             const float* __restrict__ Bsrc, int ldb,
                             float* __restrict__ C, int ldc, int Kd,
                             const float* __restrict__ bias) {
  const int lane = threadIdx.x;            // 0..31, full wave
  const int n0 = blockIdx.x * 16;
  const int m0 = blockIdx.y * 16;
  const int mn = lane & 15;
  const int kh = (lane >> 4) << 1;         // 0 or 2
  const float* arow = A + (size_t)(m0 + mn) * lda + kh;
  const float* brow = Bsrc + (size_t)(n0 + mn) * ldb + kh;
  v8f acc = {};
  for (int k0 = 0; k0 < Kd; k0 += 4) {
    v2f a, b;
    a.x = arow[k0]; a.y = arow[k0 + 1];
    b.x = brow[k0]; b.y = brow[k0 + 1];
    acc = __builtin_amdgcn_wmma_f32_16x16x4_f32(false, a, false, b,
                                                (short)0, acc, false, false);
  }
  const int mrow = (lane < 16) ? 0 : 8;
  const float bval = bias ? bias[n0 + mn] : 0.0f;
#pragma unroll
  for (int v = 0; v < 8; ++v)
    C[(size_t)(m0 + mrow + v) * ldc + n0 + mn] = acc[v] + bval;
}

// ---------------- top-8 neighbors per row (self excluded) ------------------
// One wave per row; the 256-entry row lives in 8 registers per lane.
__global__ void topk8(const float* __restrict__ sim, int* __restrict__ outIdx) {
  const int r = blockIdx.x;
  const int lane = threadIdx.x;
  float vals[8];
#pragma unroll
  for (int s = 0; s < 8; ++s) {
    const int t = lane + 32 * s;
    float v = sim[(size_t)r * BDIM + t];
    if (t == r) v = -3.0e38f;               // diag set to -1 in reference
    vals[s] = v;
  }
  for (int it = 0; it < KNN; ++it) {
    float best = -3.0e38f;
    int bidx = 1 << 30;
#pragma unroll
    for (int s = 0; s < 8; ++s) {
      const int t = lane + 32 * s;
      const float v = vals[s];
      if (v > best || (v == best && t < bidx)) { best = v; bidx = t; }
    }
    for (int off = 16; off > 0; off >>= 1) {   // wave32 reduction
      const float ov = __shfl_xor(best, off, 32);
      const int   oi = __shfl_xor(bidx, off, 32);
      if (ov > best || (ov == best && oi < bidx)) { best = ov; bidx = oi; }
    }
    if (lane == 0) outIdx[r * KNN + it] = bidx;
#pragma unroll
    for (int s = 0; s < 8; ++s)
      if (bidx == lane + 32 * s) vals[s] = -3.0e38f;  // remove winner
  }
}

// ---------------- pair MLP layer 2: h2 = relu(W2 * relu(A_i+Bm_j) + b2) ----
// 16 pairs per block staged in LDS (chunked, padded against bank conflicts),
// WMMA over the H contraction, TPW d-tiles accumulated per wave.
template <int H, int Q, int WAVES, int HC>
__global__ void __launch_bounds__(WAVES * 32)
pair_h2(const float* __restrict__ Aarr, const float* __restrict__ Bmarr,
        const int* __restrict__ topk, const float* __restrict__ W2,
        const float* __restrict__ b2, float* __restrict__ H2) {
  constexpr int TPW = Q / 16 / WAVES;
  __shared__ float hsh[16][HC + 4];          // +4 floats: 16 rows hit 16 banks
  const int p0 = blockIdx.x * 16;
  const int tid = threadIdx.x;
  const int wave = tid >> 5, lane = tid & 31;
  const int mn = lane & 15;
  const int kh = (lane >> 4) << 1;
  v8f acc[TPW] = {};
  for (int ch = 0; ch < H; ch += HC) {
    __syncthreads();
    // stage relu(A[i] + Bm[j]) for the 16 pairs (b1 pre-folded into A)
    for (int t = tid; t < 16 * HC; t += WAVES * 32) {
      const int rr = t / HC, cc = t - rr * HC;
      const int p = p0 + rr;
      const int i = p >> 3;                  // pair p = 8*i + slot
      const int j = topk[p];
      const float v = Aarr[(size_t)i * H + ch + cc] +
                      Bmarr[(size_t)j * H + ch + cc];
      hsh[rr][cc] = v > 0.f ? v : 0.f;
    }
    __syncthreads();
#pragma unroll
    for (int t2 = 0; t2 < TPW; ++t2) {
      const int d0 = (wave * TPW + t2) * 16;
      const float* wrow = W2 + (size_t)(d0 + mn) * H + ch + kh;
      v8f a_ = acc[t2];
      for (int k0 = 0; k0 < HC; k0 += 4) {
        v2f a, b;
        a.x = hsh[mn][k0 + kh]; a.y = hsh[mn][k0 + kh + 1];
        b.x = wrow[k0];         b.y = wrow[k0 + 1];
        a_ = __builtin_amdgcn_wmma_f32_16x16x4_f32(false, a, false, b,
                                                   (short)0, a_, false, false);
      }
      acc[t2] = a_;
    }
  }
  const int mrow = (lane < 16) ? 0 : 8;
#pragma unroll
  for (int t2 = 0; t2 < TPW; ++t2) {
    const int d0 = (wave * TPW + t2) * 16;
    const float bb = b2[d0 + mn];
#pragma unroll
    for (int v = 0; v < 8; ++v) {
      const float val = acc[t2][v] + bb;
      H2[(size_t)(p0 + mrow + v) * Q + d0 + mn] = val > 0.f ? val : 0.f;
    }
  }
}

// ---------------- pair head: 2-logit softmax -> score ----------------------
template <int Q>
__global__ void pair_out(const float* __restrict__ H2,
                         const float* __restrict__ w3,
                         const float* __restrict__ b3,
                         float* __restrict__ score) {
  const int p = blockIdx.x;
  const int lane = threadIdx.x;
  const float* row = H2 + (size_t)p * Q;
  float a0 = 0.f, a1 = 0.f;
  for (int d = lane; d < Q; d += 32) {
    const float h = row[d];
    a0 += h * w3[d];
    a1 += h * w3[Q + d];
  }
  for (int off = 16; off > 0; off >>= 1) {
    a0 += __shfl_xor(a0, off, 32);
    a1 += __shfl_xor(a1, off, 32);
  }
  if (lane == 0) {
    const float s0 = a0 + b3[0], s1 = a1 + b3[1];
    const float m = fmaxf(s0, s1);
    const float e0 = __expf(s0 - m), e1 = __expf(s1 - m);
    score[p] = e0 / (e0 + e1);               // softmax(...)[0]
  }
}

// ---------------- p = softmax(logits) and column sum of p ------------------
__global__ void softmax_p(const float* __restrict__ logits,
                          float* __restrict__ p, float* __restrict__ colsum) {
  const int r = threadIdx.x;                 // grid(1), block(256)
  float v[NC];
  float m = -3.0e38f;
  for (int c = 0; c < NC; ++c) { v[c] = logits[r * NC + c]; m = fmaxf(m, v[c]); }
  float s = 0.f;
  for (int c = 0; c < NC; ++c) { v[c] = __expf(v[c] - m); s += v[c]; }
  const float inv = 1.0f / s;
  for (int c = 0; c < NC; ++c) { v[c] *= inv; p[r * NC + c] = v[c]; }
  __shared__ float red[256];
  for (int c = 0; c < NC; ++c) {
    red[r] = v[c];
    __syncthreads();
    for (int off = 128; off > 0; off >>= 1) {
      if (r < off) red[r] += red[r + off];
      __syncthreads();
    }
    if (r == 0) colsum[c] = red[0];          // ones @ p
    __syncthreads();
  }
}

// ---------------- fuse masked weights, normalization, bank momentum --------
__global__ void finalize(const float* __restrict__ p,
                         const float* __restrict__ colsum,
                         const int* __restrict__ tkF, const float* __restrict__ scF,
                         const int* __restrict__ tkL, const float* __restrict__ scL,
                         const int* __restrict__ idx, const float* __restrict__ bank,
                         float* __restrict__ out) {
  const int i = threadIdx.x;                 // grid(1), block(256)
  float sf = 0.f, sl = 0.f;
  int jF[KNN], jL[KNN];
  float wF[KNN], wL[KNN];
#pragma unroll
  for (int s = 0; s < KNN; ++s) {
    jF[s] = tkF[i * KNN + s]; wF[s] = scF[i * KNN + s]; sf += wF[s];
    jL[s] = tkL[i * KNN + s]; wL[s] = scL[i * KNN + s]; sl += wL[s];
  }
  const float invF = 1.0f / (sf + EPSV);
  const float invL = 1.0f / (sl + EPSV);
  const int bi = idx[i];
  const float epsn = EPSV / (float)BDIM;
  for (int c = 0; c < NC; ++c) {
    float af = 0.f, al = 0.f;
#pragma unroll
    for (int s = 0; s < KNN; ++s) {
      af += wF[s] * p[jF[s] * NC + c];
      al += wL[s] * p[jL[s] * NC + c];
    }
    const float tf = (af + epsn * colsum[c]) * invF;  // (w + eps/n)/(sum+eps) @ p
    const float tl = (al + epsn * colsum[c]) * invL;
    const float tgt = 0.5f * (tf + tl);
    out[i * NC + c] = bank[bi * NC + c] * MOM + tgt * (1.0f - MOM);
  }
}

// ---------------------------------------------------------------------------
extern "C" void kernel_launch(void* const* d_in, const int* in_sizes, int n_in,
                              void* d_out, int out_size, void* d_ws,
                              size_t ws_size, hipStream_t stream) {
  (void)in_sizes; (void)n_in; (void)out_size; (void)ws_size;
  const float* fer_x    = (const float*)d_in[0];
  const float* lm_x     = (const float*)d_in[1];
  const float* logits   = (const float*)d_in[2];
  const int*   idx      = (const int*)d_in[3];
  const float* bank     = (const float*)d_in[4];
  const float* fer_w_in = (const float*)d_in[5];
  const float* fer_b_in = (const float*)d_in[6];
  const float* fer_w1   = (const float*)d_in[7];
  const float* fer_b1   = (const float*)d_in[8];
  const float* fer_w2   = (const float*)d_in[9];
  const float* fer_b2   = (const float*)d_in[10];
  const float* fer_w3   = (const float*)d_in[11];
  const float* fer_b3   = (const float*)d_in[12];
  const float* lm_w_in  = (const float*)d_in[13];
  const float* lm_b_in  = (const float*)d_in[14];
  const float* lm_w1    = (const float*)d_in[15];
  const float* lm_b1    = (const float*)d_in[16];
  const float* lm_w2    = (const float*)d_in[17];
  const float* lm_b2    = (const float*)d_in[18];
  const float* lm_w3    = (const float*)d_in[19];
  const float* lm_b3    = (const float*)d_in[20];

  // workspace carve-out (~12.6 MB of fp32)
  float* W = (float*)d_ws;
  float* ferN  = W; W += BDIM * FERD;
  float* lmN   = W; W += BDIM * LMD;
  float* ferS  = W; W += BDIM * BDIM;
  float* lmS   = W; W += BDIM * BDIM;
  float* ferF  = W; W += BDIM * FERD;
  float* lmF   = W; W += BDIM * LMD;
  float* ferA  = W; W += BDIM * (FERD / 2);
  float* ferB  = W; W += BDIM * (FERD / 2);
  float* lmA   = W; W += BDIM * (LMD / 2);
  float* lmB   = W; W += BDIM * (LMD / 2);
  float* ferH2 = W; W += BDIM * KNN * (FERD / 4);
  float* lmH2  = W; W += BDIM * KNN * (LMD / 4);
  float* scF   = W; W += BDIM * KNN;
  float* scL   = W; W += BDIM * KNN;
  float* pbuf  = W; W += BDIM * NC;
  float* csum  = W; W += 8;
  int* tkF = (int*)W; W += BDIM * KNN;
  int* tkL = (int*)W; W += BDIM * KNN;

  // 1) cosine-similarity neighbors
  row_normalize<<<BDIM, 256, 0, stream>>>(fer_x, ferN, FERD);
  row_normalize<<<BDIM, 256, 0, stream>>>(lm_x, lmN, LMD);
  gemm_nt_wmma<<<dim3(16, 16), 32, 0, stream>>>(ferN, FERD, ferN, FERD,
                                                ferS, BDIM, FERD, nullptr);
  gemm_nt_wmma<<<dim3(16, 16), 32, 0, stream>>>(lmN, LMD, lmN, LMD,
                                                lmS, BDIM, LMD, nullptr);
  topk8<<<BDIM, 32, 0, stream>>>(ferS, tkF);
  topk8<<<BDIM, 32, 0, stream>>>(lmS, tkL);

  // 2) feats = x @ w_in^T + b_in
  gemm_nt_wmma<<<dim3(FERD / 16, 16), 32, 0, stream>>>(
      fer_x, FERD, fer_w_in, FERD, ferF, FERD, FERD, fer_b_in);
  gemm_nt_wmma<<<dim3(LMD / 16, 16), 32, 0, stream>>>(
      lm_x, LMD, lm_w_in, LMD, lmF, LMD, LMD, lm_b_in);

  // 3) A = f1 @ w1a^T + b1 ; Bm = f2 @ w1b^T   (b1 folded into A)
  gemm_nt_wmma<<<dim3(64, 16), 32, 0, stream>>>(
      ferF, FERD, fer_w1, FERD, ferA, FERD / 2, FERD / 2, fer_b1);
  gemm_nt_wmma<<<dim3(64, 16), 32, 0, stream>>>(
      ferF + FERD / 2, FERD, fer_w1 + FERD / 2, FERD, ferB, FERD / 2,
      FERD / 2, nullptr);
  gemm_nt_wmma<<<dim3(8, 16), 32, 0, stream>>>(
      lmF, LMD, lm_w1, LMD, lmA, LMD / 2, LMD / 2, lm_b1);
  gemm_nt_wmma<<<dim3(8, 16), 32, 0, stream>>>(
      lmF + LMD / 2, LMD, lm_w1 + LMD / 2, LMD, lmB, LMD / 2, LMD / 2,
      nullptr);

  // 4) softmax of logits (+ column sum for the eps rank-1 correction)
  softmax_p<<<1, 256, 0, stream>>>(logits, pbuf, csum);

  // 5) pair MLP only at the 2048 masked (i, top-k j) positions
  pair_h2<FERD / 2, FERD / 4, 8, 512><<<BDIM * KNN / 16, 256, 0, stream>>>(
      ferA, ferB, tkF, fer_w2, fer_b2, ferH2);
  pair_h2<LMD / 2, LMD / 4, 4, LMD / 2><<<BDIM * KNN / 16, 128, 0, stream>>>(
      lmA, lmB, tkL, lm_w2, lm_b2, lmH2);
  pair_out<FERD / 4><<<BDIM * KNN, 32, 0, stream>>>(ferH2, fer_w3, fer_b3, scF);
  pair_out<LMD / 4><<<BDIM * KNN, 32, 0, stream>>>(lmH2, lm_w3, lm_b3, scL);

  // 6) normalize masked weights, mix, momentum-update against bank
  finalize<<<1, 256, 0, stream>>>(pbuf, csum, tkF, scF, tkL, scL, idx, bank,
                                  (float*)d_out);
}